// DMBERT_40200893890858
// MI455X (gfx1250) — compile-verified
//
#include <hip/hip_runtime.h>
#include <float.h>

typedef __attribute__((ext_vector_type(2))) float v2f;
typedef __attribute__((ext_vector_type(8))) float v8f;

#define BB    32
#define SS    512
#define HH    1024
#define NC    256
#define NL    34
#define NLPAD 48
#define KTOT  (2 * HH)   // 2048

// ---------------------------------------------------------------------------
// Stage 0: pad + transpose W[2048,34] -> Wt[48,2048] (cols 34..47 zeroed).
// Makes the GEMM B-fragment a single unconditional contiguous v2f load.
// ---------------------------------------------------------------------------
__global__ void dmbert_wpad_kernel(const float* __restrict__ Wm,
                                   float*       __restrict__ Wt) {
    const int idx = blockIdx.x * blockDim.x + threadIdx.x; // 0 .. 48*2048-1
    const int k   = idx & (KTOT - 1);                      // contiguous writes
    const int c   = idx / KTOT;                            // 0..47
    Wt[idx] = (c < NL) ? Wm[(size_t)k * NL + c] : 0.0f;
}

// ---------------------------------------------------------------------------
// Stage 1: per-candidate segmented max pool.
// One block per candidate, 256 threads, each owns 4 consecutive h (float4).
// Loop split at the (block-uniform) cut -> branch-free inner loops,
// fully coalesced 128-bit loads (block covers one full 4KB row per step).
// hidden (64 MB) fits in the 192 MB L2, so the 8x logical re-read per batch
// row is served from L2; HBM floor ~64MB / 23.3 TB/s.
// ---------------------------------------------------------------------------
__global__ void dmbert_pool_kernel(const float* __restrict__ hidden,
                                   const int*  __restrict__ cand_batch,
                                   const int*  __restrict__ cand_cut,
                                   float*      __restrict__ pooled) {
    const int n   = blockIdx.x;
    const int t   = threadIdx.x;           // 0..255 -> h = 4t..4t+3
    const int bat = cand_batch[n];
    const int cut = cand_cut[n];           // in [1, S-1]; uniform per block

    const float4* row = (const float4*)(hidden + (size_t)bat * SS * HH) + t;
    const int rstride = HH / 4;            // 256 float4 per token row

    float4 ml = make_float4(-FLT_MAX, -FLT_MAX, -FLT_MAX, -FLT_MAX);
    float4 mr = ml;

    int s = 0;
    #pragma unroll 4
    for (; s < cut; ++s) {
        float4 v = row[(size_t)s * rstride];
        ml.x = fmaxf(ml.x, v.x); ml.y = fmaxf(ml.y, v.y);
        ml.z = fmaxf(ml.z, v.z); ml.w = fmaxf(ml.w, v.w);
    }
    #pragma unroll 4
    for (; s < SS; ++s) {
        float4 v = row[(size_t)s * rstride];
        mr.x = fmaxf(mr.x, v.x); mr.y = fmaxf(mr.y, v.y);
        mr.z = fmaxf(mr.z, v.z); mr.w = fmaxf(mr.w, v.w);
    }

    float4* outL = (float4*)(pooled + (size_t)n * KTOT) + t;        // [0,H)
    float4* outR = (float4*)(pooled + (size_t)n * KTOT + HH) + t;   // [H,2H)
    *outL = ml;
    *outR = mr;
}

// ---------------------------------------------------------------------------
// Stage 2: logits[256,34] = pooled[256,2048] @ W[2048,34] + b
// fp32 WMMA: V_WMMA_F32_16X16X4_F32, D(16x16) = A(16x4) x B(4x16) + C.
// Grid: 16 M-tiles x 3 N-tiles (34 -> 48 padded). 4 waves/block, each wave
// owns a K-slice of 512 (128 WMMA ops), then LDS reduction across waves.
// Inner loop is now two unconditional global_load_b64 + one wmma.
// ---------------------------------------------------------------------------
__global__ void dmbert_logits_wmma_kernel(const float* __restrict__ pooled,
                                          const float* __restrict__ Wt,
                                          const float* __restrict__ bias,
                                          float*       __restrict__ out) {
    const int mtile = blockIdx.x;          // 0..15
    const int ntile = blockIdx.y;          // 0..2
    const int wave  = threadIdx.x >> 5;    // 0..3 (wave32)
    const int lane  = threadIdx.x & 31;
    const int half  = lane >> 4;           // 0: lanes 0-15, 1: lanes 16-31
    const int l16   = lane & 15;

    const int m   = mtile * 16 + l16;      // A-matrix row owned by this lane
    const int col = ntile * 16 + l16;      // B/D column owned by this lane

    // Lane-contiguous streams; K striping per ISA layout:
    //   lanes 0-15 hold (K=k, k+1), lanes 16-31 hold (K=k+2, k+3)
    const float* arow = pooled + (size_t)m   * KTOT + 2 * half;
    const float* brow = Wt     + (size_t)col * KTOT + 2 * half;

    v8f c = {};                            // fp32 accumulator tile

    const int k0 = wave * (KTOT / 4);      // 512-wide K slice per wave
    #pragma unroll 4
    for (int kk = 0; kk < KTOT / 4; kk += 4) {
        const int k = k0 + kk;
        v2f a = *(const v2f*)(arow + k);
        v2f b = *(const v2f*)(brow + k);
        c = __builtin_amdgcn_wmma_f32_16x16x4_f32(
                /*neg_a=*/false, a, /*neg_b=*/false, b,
                /*c_mod=*/(short)0, c, /*reuse_a=*/false, /*reuse_b=*/false);
    }

    // Cross-wave reduction of the partial 16x16 tiles.
    __shared__ float red[4][32][8];
    #pragma unroll
    for (int j = 0; j < 8; ++j) red[wave][lane][j] = c[j];
    __syncthreads();

    if (wave == 0 && col < NL) {
        #pragma unroll
        for (int j = 0; j < 8; ++j) {
            float s = red[0][lane][j] + red[1][lane][j] +
                      red[2][lane][j] + red[3][lane][j];
            // D layout: VGPR j, lanes 0-15 -> M=j, lanes 16-31 -> M=j+8
            const int mrow = mtile * 16 + j + half * 8;
            out[(size_t)mrow * NL + col] = s + bias[col];
        }
    }
}

extern "C" void kernel_launch(void* const* d_in, const int* in_sizes, int n_in,
                              void* d_out, int out_size, void* d_ws, size_t ws_size,
                              hipStream_t stream) {
    const float* hidden     = (const float*)d_in[0];  // [32, 512, 1024]
    const float* Wm         = (const float*)d_in[1];  // [2048, 34]
    const float* bias       = (const float*)d_in[2];  // [34]
    const int*   cand_batch = (const int*)  d_in[3];  // [256]
    const int*   cand_cut   = (const int*)  d_in[4];  // [256]
    float*       out        = (float*)d_out;          // [256, 34]

    float* pooled = (float*)d_ws;                     // [256, 2048] = 2 MB
    float* Wt     = pooled + (size_t)NC * KTOT;       // [48, 2048]  = 384 KB

    dmbert_wpad_kernel<<<(NLPAD * KTOT) / 256, 256, 0, stream>>>(Wm, Wt);

    dmbert_pool_kernel<<<NC, 256, 0, stream>>>(hidden, cand_batch, cand_cut, pooled);

    dim3 grid(NC / 16, (NL + 15) / 16);   // 16 x 3 tiles
    dmbert_logits_wmma_kernel<<<grid, 128, 0, stream>>>(pooled, Wt, bias, out);
}